// QCNN_30477087932733
// MI455X (gfx1250) — compile-verified
//
#include <hip/hip_runtime.h>
#include <hip/hip_bf16.h>

typedef __attribute__((ext_vector_type(2))) float v2f;
typedef __attribute__((ext_vector_type(8))) float v8f;

// ---------------------------------------------------------------------------
// Kernel 1: build the circuit matrix.
// Block j simulates the 8-qubit circuit applied to basis state |j>, giving
// column j of the unitary U. We keep only odd rows (wire7 == 1) and store
// A^T interleaved: AtP[(k>>1)*512 + r*2 + (k&1)] = A[r][k], where
//   r <  128 : Re(U[2r+1, k])
//   r >= 128 : Im(U[2(r-128)+1, k])
// (k = column index j of U = contraction index of the GEMM).
// ---------------------------------------------------------------------------

__device__ __forceinline__ void gate_rz(float2* st, int wire, float theta, int t) {
    float h = 0.5f * theta;
    float c = cosf(h), s = sinf(h);
    int mask = 1 << (7 - wire);
    int low  = t & (mask - 1);
    int i0   = ((t ^ low) << 1) | low;
    int i1   = i0 | mask;
    float2 a = st[i0], b = st[i1];
    // a *= exp(-i*theta/2) = (c - i s);  b *= exp(+i*theta/2) = (c + i s)
    st[i0] = make_float2(a.x * c + a.y * s, a.y * c - a.x * s);
    st[i1] = make_float2(b.x * c - b.y * s, b.y * c + b.x * s);
}

__device__ __forceinline__ void gate_ry(float2* st, int wire, float theta, int t) {
    float h = 0.5f * theta;
    float c = cosf(h), s = sinf(h);
    int mask = 1 << (7 - wire);
    int low  = t & (mask - 1);
    int i0   = ((t ^ low) << 1) | low;
    int i1   = i0 | mask;
    float2 a = st[i0], b = st[i1];
    st[i0] = make_float2(c * a.x - s * b.x, c * a.y - s * b.y);
    st[i1] = make_float2(s * a.x + c * b.x, s * a.y + c * b.y);
}

__device__ __forceinline__ void gate_cnot(float2* st, int ctrl, int tgt, int t) {
    int pc = 7 - ctrl, pt = 7 - tgt;
    int p1 = pc < pt ? pc : pt;
    int p2 = pc < pt ? pt : pc;
    if (t < 64) {
        unsigned low  = (unsigned)t & ((1u << p1) - 1u);
        unsigned mid  = ((unsigned)t >> p1) & ((1u << (p2 - p1 - 1)) - 1u);
        unsigned high = (unsigned)t >> (p2 - 1);
        unsigned i = (high << (p2 + 1)) | (mid << (p1 + 1)) | low;
        i |= (1u << pc);               // ctrl bit = 1, tgt bit = 0
        unsigned ip = i | (1u << pt);  // partner with tgt bit = 1
        float2 a = st[i], b = st[ip];
        st[i]  = b;
        st[ip] = a;
    }
}

__global__ void qcnn_build_matrix(const float* __restrict__ w, float* __restrict__ AtP) {
    __shared__ float2 st[256];
    const int j = blockIdx.x;   // basis column
    const int t = threadIdx.x;  // 128 threads

    st[t]       = make_float2((t == j)       ? 1.0f : 0.0f, 0.0f);
    st[t + 128] = make_float2((t + 128 == j) ? 1.0f : 0.0f, 0.0f);
    __syncthreads();

    const int pr1[7] = {0, 2, 4, 6, 1, 5, 3};
    const int pr2[7] = {1, 3, 5, 7, 3, 7, 7};
    const float HPI = 1.57079632679489662f;

    for (int p = 0; p < 7; ++p) {
        const int b1 = pr1[p], b2 = pr2[p];
        const float* pw = w + p * 6;
        // conv_block
        gate_rz  (st, b2, -HPI,  t); __syncthreads();
        gate_cnot(st, b2, b1,    t); __syncthreads();
        gate_rz  (st, b1, pw[0], t); __syncthreads();
        gate_ry  (st, b2, pw[1], t); __syncthreads();
        gate_cnot(st, b1, b2,    t); __syncthreads();
        gate_ry  (st, b2, pw[2], t); __syncthreads();
        gate_cnot(st, b2, b1,    t); __syncthreads();
        gate_rz  (st, b1,  HPI,  t); __syncthreads();
        // pool_block
        gate_rz  (st, b2, -HPI,  t); __syncthreads();
        gate_cnot(st, b2, b1,    t); __syncthreads();
        gate_rz  (st, b1, pw[3], t); __syncthreads();
        gate_ry  (st, b2, pw[4], t); __syncthreads();
        gate_cnot(st, b1, b2,    t); __syncthreads();
        gate_ry  (st, b2, pw[5], t); __syncthreads();
    }

    float2 amp = st[2 * t + 1];  // odd row 2t+1 of this column
    const int base = (j >> 1) * 512 + (j & 1);
    AtP[base + t * 2]         = amp.x;  // r = t       (Re)
    AtP[base + (t + 128) * 2] = amp.y;  // r = t + 128 (Im)
}

// ---------------------------------------------------------------------------
// Kernel 2: q_b = ||A x_b||^2 / ||x_b||^2 via fp32 WMMA GEMM.
// Block: 512 threads (16 waves), 128 batch rows. Wave w owns output columns
// [16w, 16w+16); accumulates 8 stacked 16x16 C tiles over K=256 (step 4).
// Each B fragment (one global b64 from L2-resident At) feeds 8 WMMAs.
// ---------------------------------------------------------------------------

#define BM 128
#define NSUB 8
#define NWAVES 16

__global__ __launch_bounds__(512) void qcnn_wmma_gemm(const float* __restrict__ x,
                                                      const float2* __restrict__ AtP,
                                                      float* __restrict__ out) {
    __shared__ float Xs[BM * 260];        // padded: stride 260 -> conflict-free
    __shared__ float part[BM][NWAVES];    // per-wave row sum-of-squares partials
    __shared__ float xss[BM][4];          // per-row ||x||^2 partials

    const int tid = threadIdx.x;
    const size_t m0 = (size_t)blockIdx.x * BM;

    // Stage X tile into LDS and accumulate ||x||^2 partials.
    {
        const int row = tid >> 2;  // 0..127 (4 threads per row)
        const int seg = tid & 3;   // 4 segs x 64 cols
        const float4* src = (const float4*)(x + (m0 + row) * 256 + seg * 64);
        float* dst = &Xs[row * 260 + seg * 64];
        float ss = 0.0f;
#pragma unroll
        for (int i = 0; i < 16; ++i) {
            float4 v = src[i];
            dst[i * 4 + 0] = v.x; dst[i * 4 + 1] = v.y;
            dst[i * 4 + 2] = v.z; dst[i * 4 + 3] = v.w;
            ss += v.x * v.x + v.y * v.y + v.z * v.z + v.w * v.w;
        }
        xss[row][seg] = ss;
    }
    __syncthreads();

    const int wave  = tid >> 5;
    const int lane  = tid & 31;
    const int laneL = lane & 15;
    const int hi    = lane >> 4;   // half-wave selects K pair (k,k+1) vs (k+2,k+3)
    const int n0    = wave * 16;

    v8f acc[NSUB] = {};

#pragma unroll 4
    for (int k = 0; k < 256; k += 4) {
        // B fragment: v0 = At[k+2*hi][n], v1 = At[k+1+2*hi][n] (one b64 load)
        float2 bv = AtP[(size_t)((k >> 1) + hi) * 256 + n0 + laneL];
        v2f b; b[0] = bv.x; b[1] = bv.y;
#pragma unroll
        for (int s = 0; s < NSUB; ++s) {
            // A fragment from LDS: v0 = X[m][k+2*hi], v1 = X[m][k+1+2*hi]
            const float2* pr = (const float2*)&Xs[(s * 16 + laneL) * 260];
            float2 av = pr[(k >> 1) + hi];
            v2f a; a[0] = av.x; a[1] = av.y;
            acc[s] = __builtin_amdgcn_wmma_f32_16x16x4_f32(
                false, a, false, b, (short)0, acc[s], false, false);
        }
    }

    // Per-row sum of C^2 over this wave's 16 columns.
    // C layout: vgpr v, lanes 0-15 -> row v; lanes 16-31 -> row v+8.
#pragma unroll
    for (int s = 0; s < NSUB; ++s) {
#pragma unroll
        for (int v = 0; v < 8; ++v) {
            float r2 = acc[s][v] * acc[s][v];
            r2 += __shfl_xor(r2, 1);
            r2 += __shfl_xor(r2, 2);
            r2 += __shfl_xor(r2, 4);
            r2 += __shfl_xor(r2, 8);   // masks < 16: halves reduce independently
            if (laneL == 0) part[s * 16 + hi * 8 + v][wave] = r2;
        }
    }
    __syncthreads();

    if (tid < BM) {
        float q = 0.0f;
#pragma unroll
        for (int wv = 0; wv < NWAVES; ++wv) q += part[tid][wv];
        float ssx = 0.0f;
#pragma unroll
        for (int i = 0; i < 4; ++i) ssx += xss[tid][i];
        q /= ssx;
        const size_t bidx = m0 + tid;
        out[2 * bidx + 0] = 1.0f - q;
        out[2 * bidx + 1] = q;
    }
}

// ---------------------------------------------------------------------------

extern "C" void kernel_launch(void* const* d_in, const int* in_sizes, int n_in,
                              void* d_out, int out_size, void* d_ws, size_t ws_size,
                              hipStream_t stream) {
    const float* x = (const float*)d_in[0];       // (131072, 256) f32
    const float* w = (const float*)d_in[1];       // (42,) f32
    float* out = (float*)d_out;                   // (131072, 2) f32
    float* AtP = (float*)d_ws;                    // 256*256 f32 = 256 KB scratch

    const int B = in_sizes[0] / 256;

    qcnn_build_matrix<<<256, 128, 0, stream>>>(w, AtP);
    qcnn_wmma_gemm<<<B / BM, 512, 0, stream>>>(x, (const float2*)AtP, out);
}